// Embedding_78795470013070
// MI455X (gfx1250) — compile-verified
//
#include <hip/hip_runtime.h>

// Embedding gather: out[i, :] = W[idx[i], :], D = 128 floats (512 B per row).
// Pure-bandwidth kernel: ~2.06 GB of traffic, floor ~88 us at 23.3 TB/s HBM.
//
// One wave (32 lanes) copies one row per load/store pair: lane l moves float4
// number l (32 lanes * 16 B = 512 B = full row, 8 whole 64 B cachelines, zero
// read waste). Each wave owns 32 consecutive output rows:
//   - ONE coalesced b32 load fetches the wave's 32 indices (lane l holds row
//     base+l's index).
//   - Each row's index is broadcast with v_readlane_b32 (constant lane) into
//     an SGPR, so the gather address is uniform: SALU address math + saddr-form
//     global_load_b128 with lane*16 as the only vector offset.
//   - Stores reuse a single per-wave address VGPR; the constant 512 B per-row
//     offsets fold into the 24-bit instruction offset.
//
// Cache policy: table loads regular-temporal (2M lookups over 1M rows => ~2x
// reuse; 192 MB L2 holds a good fraction of the 512 MB table). Output stores
// non-temporal (written once, never read) so the 1 GB stream doesn't evict
// table lines. Index load non-temporal (8 MB, read once).

typedef float v4f __attribute__((ext_vector_type(4)));

#define WAVES_PER_BLOCK 8        // 256 threads = 8 wave32
#define ROWS_PER_WAVE   32       // one coalesced index load per wave
#define UNROLL          8        // 8 b128 loads in flight per wave per group

__global__ __launch_bounds__(256) void emb_gather_kernel(
    const v4f* __restrict__ W,      // table as float4[N_EMB * 32]
    const int* __restrict__ idx,    // row indices
    v4f*       __restrict__ out,    // output as float4[n_rows * 32]
    int n_rows)
{
    const int lane = threadIdx.x & 31;
    const long long waveId =
        (((long long)blockIdx.x * blockDim.x) + threadIdx.x) >> 5;
    const long long rowBase = waveId * ROWS_PER_WAVE;
    if (rowBase >= n_rows) return;

    if (rowBase + ROWS_PER_WAVE <= n_rows) {
        // Fast path: one coalesced NT load grabs this wave's 32 row indices.
        const int laneRowIdx = __builtin_nontemporal_load(&idx[rowBase + lane]);

        // Single per-wave output pointer; per-row offsets are compile-time
        // constants that fold into the instruction's immediate offset.
        v4f* const o = out + (size_t)rowBase * 32 + lane;

        #pragma unroll
        for (int j = 0; j < ROWS_PER_WAVE; j += UNROLL) {
            v4f v[UNROLL];
            #pragma unroll
            for (int k = 0; k < UNROLL; ++k) {
                // v_readlane_b32 -> SGPR row index -> uniform (SALU) address
                // -> saddr-form global_load_b128.
                const int r = __builtin_amdgcn_readlane(laneRowIdx, j + k);
                const v4f* const rowPtr = W + (size_t)r * 32;
                v[k] = rowPtr[lane];
            }
            #pragma unroll
            for (int k = 0; k < UNROLL; ++k) {
                // Streaming store: th:TH_STORE_NT, keep L2 for the table.
                __builtin_nontemporal_store(v[k], o + (size_t)(j + k) * 32);
            }
        }
    } else {
        // Tail path (unused when n_rows % 32 == 0, e.g. the 2^21 reference).
        for (long long row = rowBase; row < n_rows; ++row) {
            const int r = idx[row];
            v4f v = W[(size_t)r * 32 + lane];
            __builtin_nontemporal_store(v, &out[(size_t)row * 32 + lane]);
        }
    }
}

extern "C" void kernel_launch(void* const* d_in, const int* in_sizes, int n_in,
                              void* d_out, int out_size, void* d_ws, size_t ws_size,
                              hipStream_t stream)
{
    // setup_inputs() order: weight (float32, 1M x 128), index (int32, 2M)
    const v4f* W   = (const v4f*)d_in[0];
    const int* idx = (const int*)d_in[1];
    v4f*       out = (v4f*)d_out;

    const int n_rows = in_sizes[1];                    // number of lookups
    const int rows_per_block = WAVES_PER_BLOCK * ROWS_PER_WAVE;  // 256 rows
    const int blocks = (n_rows + rows_per_block - 1) / rows_per_block;

    emb_gather_kernel<<<blocks, WAVES_PER_BLOCK * 32, 0, stream>>>(
        W, idx, out, n_rows);
}